// GNN_46703474377009
// MI455X (gfx1250) — compile-verified
//
#include <hip/hip_runtime.h>
#include <math.h>

// ---------------------------------------------------------------------------
// RWKP-GNN forward for MI455X (gfx1250), wave32, WMMA f32 16x16x4.
// N=50000 nodes, E=800000 edges, 64->128 hidden, 3 layers, MAX_STEP=2.
// All dense GEMMs go through V_WMMA_F32_16X16X4_F32 (exact fp32).
// B panel is pair-packed in LDS so every WMMA B fragment is one ds_load_b64.
// ---------------------------------------------------------------------------

typedef float v2f __attribute__((ext_vector_type(2)));
typedef float v8f __attribute__((ext_vector_type(8)));

#define NH128 128
#define BN_EPS 1e-5f

enum { EP_NONE = 0, EP_BIAS = 1, EP_SIGMOID = 2, EP_MUL = 3, EP_MUL2 = 4 };

// ---------------------------------------------------------------------------
// GEMM: D[nrows x 128] = A[nrows x K] * B[K x 128]  (+ fused epilogue)
// Block = 256 threads = 8 waves; block tile 64 rows x 128 cols.
// Wave w: row-tile rt = (w&3)*16, col-half n0 = (w>>2)*64 -> 4 16x16 C tiles.
// WMMA f32 16x16x4 fragment layouts (wave32):
//   A 16x4 : reg r, lane half h, m = lane&15 -> A[m][k0 + r + 2h]
//   B 4x16 : reg r, half h, n = lane&15     -> B[k0 + r + 2h][n]
//   C 16x16: reg r, half h                  -> C[r + 8h][lane&15]
// kb = k0 + 2*half is even, so (B[kb][n], B[kb+1][n]) is exactly one packed
// pair in sB2 -> single 8B LDS load per fragment, even-register aligned.
// ---------------------------------------------------------------------------
template <int K>
__global__ __launch_bounds__(256) void gemm_wmma_k(
    const float* __restrict__ A, const float* __restrict__ B,
    float* __restrict__ D, const float* __restrict__ aux, int nrows, int ep) {
  __shared__ v2f sB2[(K / 2) * NH128];  // [k/2][n] = (B[k][n], B[k+1][n])
  __shared__ float sA[64][K + 4];

  const int tid  = threadIdx.x;
  const int wave = tid >> 5;
  const int lane = tid & 31;
  const int half = lane >> 4;
  const int l16  = lane & 15;
  const int m0   = blockIdx.x * 64;

  // Stage B panel pair-packed. i = kp*128 + n; reads two contiguous rows of B
  // per 128-thread group (fully coalesced), writes one float2 to LDS.
  for (int i = tid; i < (K / 2) * NH128; i += 256) {
    int kp = i >> 7;         // k pair index
    int n  = i & (NH128 - 1);
    v2f p;
    p.x = B[(2 * kp) * NH128 + n];
    p.y = B[(2 * kp + 1) * NH128 + n];
    sB2[i] = p;
  }

  // Stage A tile (64 x K), zero-padded past nrows. lda == K for every GEMM.
  for (int i = tid; i < 64 * (K / 4); i += 256) {
    int r  = i / (K / 4);
    int c  = (i % (K / 4)) * 4;
    int gr = m0 + r;
    float4 v = make_float4(0.f, 0.f, 0.f, 0.f);
    if (gr < nrows) v = *(const float4*)(A + (size_t)gr * K + c);
    *(float4*)(&sA[r][c]) = v;
  }
  __syncthreads();

  const int n0 = (wave >> 2) * 64;
  const int rt = (wave & 3) * 16;

  v8f acc[4] = {};
  for (int k0 = 0; k0 < K; k0 += 4) {
    const int kb = k0 + 2 * half;              // even
    v2f a = *(const v2f*)&sA[rt + l16][kb];    // contiguous (A[m][kb],A[m][kb+1])
#pragma unroll
    for (int t = 0; t < 4; ++t) {
      v2f b = sB2[(kb >> 1) * NH128 + n0 + t * 16 + l16];  // one ds_load_b64
      acc[t] = __builtin_amdgcn_wmma_f32_16x16x4_f32(
          false, a, false, b, (short)0, acc[t], false, false);
    }
  }

#pragma unroll
  for (int t = 0; t < 4; ++t) {
#pragma unroll
    for (int r = 0; r < 8; ++r) {
      int gm = m0 + rt + r + 8 * half;
      int gn = n0 + t * 16 + l16;
      if (gm < nrows) {
        float v = acc[t][r];
        if (ep == EP_BIAS) {
          v += aux[gn];
        } else if (ep == EP_SIGMOID) {
          v = 1.f / (1.f + expf(-v));
        } else if (ep == EP_MUL) {
          v *= aux[(size_t)gm * NH128 + gn];
        } else if (ep == EP_MUL2) {
          float m = aux[(size_t)gm * NH128 + gn];
          v *= m * m;
        }
        D[(size_t)gm * NH128 + gn] = v;
      }
    }
  }
}

// ------------------------- adjacency normalization -------------------------
__global__ void deg_init_k(float* d, int n) {
  int i = blockIdx.x * blockDim.x + threadIdx.x;
  if (i < n) d[i] = 1.f;  // self-loop weight
}
__global__ void deg_accum_k(const int* __restrict__ row,
                            const int* __restrict__ col, float* d, int E) {
  int e = blockIdx.x * blockDim.x + threadIdx.x;
  if (e < E) {
    int r = row[e], c = col[e];
    if (r != c) atomicAdd(&d[c], 1.f);
  }
}
__global__ void deg_fin_k(float* d, int n) {  // deg -> dinv in place
  int i = blockIdx.x * blockDim.x + threadIdx.x;
  if (i < n) {
    float v = d[i];
    d[i] = (v > 0.f) ? rsqrtf(v) : 0.f;
  }
}
__global__ void edgew_k(const int* __restrict__ row, const int* __restrict__ col,
                        const float* __restrict__ dinv, float* ew, int E) {
  int e = blockIdx.x * blockDim.x + threadIdx.x;
  if (e < E) {
    int r = row[e], c = col[e];
    ew[e] = (r != c) ? dinv[r] * dinv[c] : 0.f;
  }
}

// ---------------------------------- SpMM -----------------------------------
// T[i][:] = dinv[i]^2 * Y[i][:]   (folded self-loops)
__global__ void spmm_init_k(const float* __restrict__ dinv,
                            const float* __restrict__ Y, float* T, int total) {
  int i = blockIdx.x * blockDim.x + threadIdx.x;
  if (i < total) {
    float d = dinv[i >> 7];
    T[i] = d * d * Y[i];
  }
}
// T[row[e]][:] += w[e] * Y[col[e]][:]   (one wave32 per edge)
__global__ __launch_bounds__(256) void spmm_edge_k(
    const int* __restrict__ row, const int* __restrict__ col,
    const float* __restrict__ ew, const float* __restrict__ Y, float* T, int E) {
  int g = blockIdx.x * blockDim.x + threadIdx.x;
  int e = g >> 5;
  int lane = g & 31;
  if (e >= E) return;
  float w = ew[e];
  if (w == 0.f) return;  // self loop
  const float* y = Y + (size_t)col[e] * NH128;
  float* t = T + (size_t)row[e] * NH128;
#pragma unroll
  for (int i = 0; i < 4; ++i)
    atomicAdd(t + lane + i * 32, w * y[lane + i * 32]);
}

// -------------------------------- BatchNorm --------------------------------
__global__ void stats_init_k(float* s) {
  int i = blockIdx.x * blockDim.x + threadIdx.x;
  if (i < 256) s[i] = 0.f;
}
__global__ __launch_bounds__(128) void stats_accum_k(const float* __restrict__ H,
                                                     float* s, int n) {
  int c = threadIdx.x;  // column owner
  float sum = 0.f, ss = 0.f;
  for (int r = blockIdx.x; r < n; r += gridDim.x) {
    float v = H[(size_t)r * NH128 + c];
    sum += v;
    ss += v * v;
  }
  atomicAdd(&s[c], sum);
  atomicAdd(&s[NH128 + c], ss);
}
__global__ void stats_final_k(const float* __restrict__ s,
                              const float* __restrict__ g,
                              const float* __restrict__ bt, float* sc, int n) {
  int c = threadIdx.x;
  if (c < NH128) {
    float inv = 1.f / (float)n;
    float m = s[c] * inv;
    float var = s[NH128 + c] * inv - m * m;
    float scale = g[c] * rsqrtf(var + BN_EPS);
    sc[c] = scale;
    sc[NH128 + c] = bt[c] - m * scale;
  }
}
__global__ void bn_relu_k(const float* __restrict__ H, const float* __restrict__ sc,
                          float* O, int total) {
  int i = blockIdx.x * blockDim.x + threadIdx.x;
  if (i < total) {
    int c = i & (NH128 - 1);
    O[i] = fmaxf(sc[c] * H[i] + sc[NH128 + c], 0.f);
  }
}
__global__ void bn_relu_add_k(const float* __restrict__ H,
                              const float* __restrict__ sc,
                              const float* __restrict__ P, float* O, int total) {
  int i = blockIdx.x * blockDim.x + threadIdx.x;
  if (i < total) {
    int c = i & (NH128 - 1);
    O[i] = fmaxf(sc[c] * H[i] + sc[NH128 + c], 0.f) + P[i];
  }
}

// ------------------------------ orchestration ------------------------------
extern "C" void kernel_launch(void* const* d_in, const int* in_sizes, int n_in,
                              void* d_out, int out_size, void* d_ws,
                              size_t ws_size, hipStream_t stream) {
  const float* x     = (const float*)d_in[0];
  const int*   ei    = (const int*)d_in[1];  // [2,E]
  const float* W_in  = (const float*)d_in[2];
  const float* b_in  = (const float*)d_in[3];
  const float* g_in  = (const float*)d_in[4];
  const float* bt_in = (const float*)d_in[5];
  const float* Wf    = (const float*)d_in[6];
  const float* Wa    = (const float*)d_in[7];
  const float* g_n   = (const float*)d_in[8];
  const float* bt_n  = (const float*)d_in[9];
  const float* W_o1  = (const float*)d_in[10];
  const float* b_o1  = (const float*)d_in[11];
  const float* g_o   = (const float*)d_in[12];
  const float* bt_o  = (const float*)d_in[13];
  const float* W_o2  = (const float*)d_in[14];
  const float* b_o2  = (const float*)d_in[15];

  const int N = in_sizes[0] / 64;
  const int E = in_sizes[1] / 2;
  const int* row = ei;
  const int* col = ei + E;
  const size_t NHf = (size_t)N * NH128;
  const int totalNH = (int)NHf;

  // Workspace layout (floats). Total ~131 MB for N=50000, E=800000.
  float* W = (float*)d_ws;
  size_t o = 0;
  float* dinv  = W + o; o += ((size_t)N + 255) & ~(size_t)255;
  float* ew    = W + o; o += ((size_t)E + 255) & ~(size_t)255;
  float* stats = W + o; o += 256;  // sum[128], sumsq[128]
  float* sc    = W + o; o += 256;  // scale[128], shift[128]
  float* S0    = W + o; o += NHf;  // stable h / prev (ping)
  float* S1    = W + o; o += NHf;  // stable h / prev (pong)
  float* y0    = W + o; o += NHf;  // sigmoid gate
  float* t     = W + o; o += NHf;  // SpMM accumulator / GEMM staging
  float* yb    = W + o; o += NHf;  // conv output
  (void)ws_size; (void)n_in; (void)out_size;

  const int gN    = (N + 255) / 256;
  const int gE    = (E + 255) / 256;
  const int gNH   = (totalNH + 255) / 256;
  const int gGemm = (N + 63) / 64;
  const int gEdge = (int)(((size_t)E * 32 + 255) / 256);

  // --- symmetric adjacency normalization (self-loops folded) ---
  deg_init_k<<<gN, 256, 0, stream>>>(dinv, N);
  deg_accum_k<<<gE, 256, 0, stream>>>(row, col, dinv, E);
  deg_fin_k<<<gN, 256, 0, stream>>>(dinv, N);
  edgew_k<<<gE, 256, 0, stream>>>(row, col, dinv, ew, E);

  // --- input encoder: relu(BN(x @ W_in + b_in)) ---
  gemm_wmma_k<64><<<gGemm, 256, 0, stream>>>(x, W_in, t, b_in, N, EP_BIAS);
  stats_init_k<<<1, 256, 0, stream>>>(stats);
  stats_accum_k<<<512, 128, 0, stream>>>(t, stats, N);
  stats_final_k<<<1, 128, 0, stream>>>(stats, g_in, bt_in, sc, N);
  bn_relu_k<<<gNH, 256, 0, stream>>>(t, sc, S0, totalNH);

  float* h = S0;      // current h == prev (reference invariant at layer entry)
  float* other = S1;
  for (int l = 0; l < 3; ++l) {
    const float* Wf_l = Wf + (size_t)l * NH128 * NH128;
    const float* Wa_l = Wa + (size_t)l * NH128 * NH128;
    // y0 = sigmoid(h @ Wf)
    gemm_wmma_k<128><<<gGemm, 256, 0, stream>>>(h, Wf_l, y0, nullptr, N, EP_SIGMOID);
    // step 0: y = y0^2 * (spmm(y0) @ Wa)
    spmm_init_k<<<gNH, 256, 0, stream>>>(dinv, y0, t, totalNH);
    spmm_edge_k<<<gEdge, 256, 0, stream>>>(row, col, ew, y0, t, E);
    gemm_wmma_k<128><<<gGemm, 256, 0, stream>>>(t, Wa_l, yb, y0, N, EP_MUL2);
    // step 1: y = y0 * (spmm(y) @ Wa)
    spmm_init_k<<<gNH, 256, 0, stream>>>(dinv, yb, t, totalNH);
    spmm_edge_k<<<gEdge, 256, 0, stream>>>(row, col, ew, yb, t, E);
    gemm_wmma_k<128><<<gGemm, 256, 0, stream>>>(t, Wa_l, yb, y0, N, EP_MUL);
    if (l < 2) {
      // h = relu(BN(y)) + prev ; prev = h
      stats_init_k<<<1, 256, 0, stream>>>(stats);
      stats_accum_k<<<512, 128, 0, stream>>>(yb, stats, N);
      stats_final_k<<<1, 128, 0, stream>>>(stats, g_n + l * NH128,
                                           bt_n + l * NH128, sc, N);
      bn_relu_add_k<<<gNH, 256, 0, stream>>>(yb, sc, h, other, totalNH);
      float* tmp = h; h = other; other = tmp;
    }
  }

  // --- output encoder: relu(BN(yb @ W_o1 + b_o1)) @ W_o2 + b_o2 ---
  gemm_wmma_k<128><<<gGemm, 256, 0, stream>>>(yb, W_o1, t, b_o1, N, EP_BIAS);
  stats_init_k<<<1, 256, 0, stream>>>(stats);
  stats_accum_k<<<512, 128, 0, stream>>>(t, stats, N);
  stats_final_k<<<1, 128, 0, stream>>>(stats, g_o, bt_o, sc, N);
  bn_relu_k<<<gNH, 256, 0, stream>>>(t, sc, y0, totalNH);
  gemm_wmma_k<128><<<gGemm, 256, 0, stream>>>(y0, W_o2, (float*)d_out, b_o2,
                                              N, EP_BIAS);
}